// EdgeInvariantPointAttention_60541859004906
// MI455X (gfx1250) — compile-verified
//
#include <hip/hip_runtime.h>
#include <math.h>

// ---------------- problem constants (from reference) ----------------
constexpr int NN  = 4096;   // N nodes
constexpr int KK  = 32;     // neighbors
constexpr int C_N = 384;    // node channels
constexpr int C_Z = 128;    // pair channels
constexpr int C_H = 16;     // head channels
constexpr int NH  = 12;     // heads
constexpr int NPQ = 4;      // query points
constexpr int NPV = 8;      // value points
constexpr int CATW = NH * (C_Z + C_H + NPV * 4); // 2112
constexpr int PROJW = 1152; // 192 (q) + 384 (kv) + 144 (qp) + 432 (kvp)
constexpr int OFF_Q = 0, OFF_KV = 192, OFF_QP = 576, OFF_KVP = 720;
constexpr float INF_ = 100000000.0f;
constexpr float EPS_ = 1e-6f;

typedef __attribute__((ext_vector_type(2))) float v2f;
typedef __attribute__((ext_vector_type(8))) float v8f;

__device__ __forceinline__ v8f wmma4(v2f a, v2f b, v8f c) {
  // V_WMMA_F32_16X16X4_F32: exact fp32, D = A(16x4) * B(4x16) + C
  return __builtin_amdgcn_wmma_f32_16x16x4_f32(
      /*neg_a=*/false, a, /*neg_b=*/false, b,
      /*c_mod=*/(short)0, c, /*reuse_a=*/false, /*reuse_b=*/false);
}

// ---------------------------------------------------------------
// Pack [W_q | W_kv | W_qp | W_kvp] -> Wcat [C_N, 1152]
// ---------------------------------------------------------------
__global__ __launch_bounds__(256) void pack_w(
    const float* __restrict__ W_q, const float* __restrict__ W_kv,
    const float* __restrict__ W_qp, const float* __restrict__ W_kvp,
    float* __restrict__ Wcat)
{
  const int i = blockIdx.x * 256 + threadIdx.x;
  if (i >= C_N * PROJW) return;
  const int k = i / PROJW, c = i % PROJW;
  float v;
  if      (c < OFF_KV)  v = W_q  [k * 192 + c];
  else if (c < OFF_QP)  v = W_kv [k * 384 + (c - OFF_KV)];
  else if (c < OFF_KVP) v = W_qp [k * 144 + (c - OFF_QP)];
  else                  v = W_kvp[k * 432 + (c - OFF_KVP)];
  Wcat[i] = v;
}

// ---------------------------------------------------------------
// LDS-tiled fp32 GEMM via V_WMMA_F32_16X16X4_F32.
// C[M,N] = A[M,K] * B[K,N] (+bias). Block = 64x64 outputs, 4 waves;
// wave w owns rows 16w..16w+15 and all 64 cols (4 v8f accumulators).
// Requires M%64==0, N%64==0, K%32==0.
// Operand layouts (ISA 7.12.2): A 16x4 f32: lane(0-15)=M, VGPR{0,1}=K{0,1};
// lanes 16-31: K{2,3}. B 4x16: lane=N, VGPR r = K r+2*half.
// D: VGPR r -> M = r + 8*half.
// B is staged K-pair-interleaved so a lane's {K,K+1} fragment is ONE
// aligned ds_load_b64 (no cross-row packing movs in the hot loop).
// ---------------------------------------------------------------
constexpr int KC  = 32;  // K chunk staged in LDS
constexpr int AP  = 36;  // padded A row (floats): keeps 16B align, avoids bank conflicts

__global__ __launch_bounds__(128) void gemm_wmma_f32(
    const float* __restrict__ A, const float* __restrict__ Bm,
    const float* __restrict__ bias, float* __restrict__ C,
    int M, int Kd, int Nd)
{
  const int nBlkN = Nd >> 6;
  const int bm = blockIdx.x / nBlkN;
  const int bn = blockIdx.x % nBlkN;
  const int tid  = threadIdx.x;
  const int wave = tid >> 5;
  const int lane = tid & 31;
  const int half = lane >> 4;
  const int l15  = lane & 15;
  const int rowBase = bm * 64;
  const int colBase = bn * 64;

  __shared__ float As[64][AP];        // 9 KB (padded rows -> b64 frags, no conflicts)
  __shared__ float BsP[KC / 2][64][2];// 8 KB, K-pair interleaved: [kp][n][k&1]

  v8f acc0 = {0.f,0.f,0.f,0.f,0.f,0.f,0.f,0.f};
  v8f acc1 = acc0, acc2 = acc0, acc3 = acc0;

  for (int kc = 0; kc < Kd; kc += KC) {
    // stage A chunk: 64 rows x 32 K, coalesced float4 loads (512 float4, 4/thread)
    for (int i = tid; i < 64 * (KC / 4); i += 128) {
      const int r = i >> 3, c4 = i & 7;
      const float4 v = *(const float4*)(A + (size_t)(rowBase + r) * Kd + kc + c4 * 4);
      *(float4*)&As[r][c4 * 4] = v;
    }
    // stage B chunk: 16 K-pairs x 64 cols. Unit = (row-pair, 4 cols):
    // 2 coalesced float4 global loads -> 4 adjacent v2f pairs (32 contig bytes).
    for (int u = tid; u < (KC / 2) * (64 / 4); u += 128) {
      const int rp = u >> 4;            // K-pair index 0..15
      const int c  = (u & 15) * 4;      // column group
      const float* b0p = Bm + (size_t)(kc + rp * 2) * Nd + colBase + c;
      const float4 r0 = *(const float4*)(b0p);
      const float4 r1 = *(const float4*)(b0p + Nd);
      v2f p0; p0.x = r0.x; p0.y = r1.x;
      v2f p1; p1.x = r0.y; p1.y = r1.y;
      v2f p2; p2.x = r0.z; p2.y = r1.z;
      v2f p3; p3.x = r0.w; p3.y = r1.w;
      *(v2f*)&BsP[rp][c + 0][0] = p0;
      *(v2f*)&BsP[rp][c + 1][0] = p1;
      *(v2f*)&BsP[rp][c + 2][0] = p2;
      *(v2f*)&BsP[rp][c + 3][0] = p3;
    }
    __syncthreads();

    // prefetch next chunk while we compute this one (global_prefetch_b8)
    if (kc + KC < Kd) {
      __builtin_prefetch(A + (size_t)(rowBase + (tid & 63)) * Kd + kc + KC + (tid >> 6) * 16, 0, 3);
      __builtin_prefetch(Bm + (size_t)(kc + KC + (tid & 31)) * Nd + colBase + (tid >> 5) * 16, 0, 3);
    }

    #pragma unroll
    for (int k0 = 0; k0 < KC; k0 += 4) {
      const int kp = (k0 >> 1) + half;   // pair holding rows {k0+2*half, +1}
      v2f av;
      av.x = As[wave * 16 + l15][k0 + 2 * half];
      av.y = As[wave * 16 + l15][k0 + 2 * half + 1];
      const v2f b0 = *(const v2f*)&BsP[kp][ 0 + l15][0];
      const v2f b1 = *(const v2f*)&BsP[kp][16 + l15][0];
      const v2f b2 = *(const v2f*)&BsP[kp][32 + l15][0];
      const v2f b3 = *(const v2f*)&BsP[kp][48 + l15][0];
      acc0 = wmma4(av, b0, acc0);
      acc1 = wmma4(av, b1, acc1);
      acc2 = wmma4(av, b2, acc2);
      acc3 = wmma4(av, b3, acc3);
    }
    __syncthreads();
  }

  // store D: row = rowBase + wave*16 + half*8 + r, col = colBase + j*16 + l15
  const int orow = rowBase + wave * 16 + half * 8;
  #pragma unroll
  for (int j = 0; j < 4; ++j) {
    const v8f& a = (j == 0) ? acc0 : (j == 1) ? acc1 : (j == 2) ? acc2 : acc3;
    const int col = colBase + j * 16 + l15;
    const float badd = bias ? bias[col] : 0.0f;
    #pragma unroll
    for (int r = 0; r < 8; ++r)
      C[(size_t)(orow + r) * Nd + col] = a[r] + badd;
  }
}

// ---------------------------------------------------------------
// Per-node IPA. One block (128 threads / 4 waves) per node n.
// proj row layout: [q(192) | kv(384) | qp(144) | kvp(432)], stride 1152.
// cat layout: [o(192) | opf_x(96) | opf_y(96) | opf_z(96) | norm(96) | o_pair(1536)]
// o_pair = probs[16(pad)x32] @ z[32x128] computed with WMMA from LDS.
// ---------------------------------------------------------------
__global__ __launch_bounds__(128) void ipa_attn(
    const float* __restrict__ proj,      // [N, 1152]
    const float* __restrict__ z_e,       // [N, K, 128]
    const float* __restrict__ direction, // [N, K, 3]
    const float* __restrict__ rel_ori,   // [N, K, 3, 3]
    const float* __restrict__ pair_mask, // [N, K]
    const int*   __restrict__ E_idx,     // [N, K]
    const float* __restrict__ W_b,       // [128, 12]
    const float* __restrict__ head_weights, // [12]
    float* __restrict__ cat)             // [N, 2112]
{
  const int n   = blockIdx.x;
  const int tid = threadIdx.x;

  __shared__ int   e_sh[KK];
  __shared__ float z_sh[KK][C_Z];            // 16 KB  (B operand for o_pair WMMA)
  __shared__ float vpts_sh[KK][NH][NPV][3];  // 36 KB
  __shared__ float prob[16][KK + 4];         // padded to 16 heads, 8B-aligned rows
  __shared__ float q_sh[NH * C_H];
  __shared__ float qpts_sh[3 * NH * NPQ];
  __shared__ float hw_sh[NH];
  __shared__ float opt_sh[NH][NPV][3];

  if (tid < KK) e_sh[tid] = E_idx[(size_t)n * KK + tid];
  // zero pad rows 12..14 and row 15 cols 0..19 of prob (128 entries, one per thread)
  {
    const int r = 12 + tid / (KK + 4), c = tid % (KK + 4);
    if (r < 16) prob[r][c] = 0.f;
  }
  for (int i = tid; i < NH * C_H; i += 128)
    q_sh[i] = proj[(size_t)n * PROJW + OFF_Q + i];
  for (int i = tid; i < 3 * NH * NPQ; i += 128)
    qpts_sh[i] = proj[(size_t)n * PROJW + OFF_QP + i];
  for (int i = tid; i < KK * C_Z; i += 128)
    z_sh[i >> 7][i & 127] = z_e[(size_t)n * KK * C_Z + i];
  if (tid < NH) {
    const float x = head_weights[tid];
    hw_sh[tid] = log1pf(expf(x)) * 0.13608276348795434f; // softplus * sqrt(1/54)
  }
  __syncthreads();

  const float s_qk = 0.14433756729740643f; // sqrt(1/(3*CH))
  const float s_b  = 0.5773502691896258f;  // sqrt(1/3)

  // ---- logits over (k, h) ----
  for (int t = tid; t < KK * NH; t += 128) {
    const int k = t / NH, h = t % NH;
    const int e = e_sh[k];
    const float* R = rel_ori   + (size_t)(n * KK + k) * 9;
    const float* D = direction + (size_t)(n * KK + k) * 3;
    const float R00=R[0],R01=R[1],R02=R[2],R10=R[3],R11=R[4],R12=R[5],R20=R[6],R21=R[7],R22=R[8];
    const float dx=D[0], dy=D[1], dz=D[2];
    const float* pts = proj + (size_t)e * PROJW + OFF_KVP + h * (NPQ + NPV);
    float ptsum = 0.f;
    #pragma unroll
    for (int p = 0; p < NPQ + NPV; ++p) {
      const float px = pts[0 * NH * (NPQ + NPV) + p];
      const float py = pts[1 * NH * (NPQ + NPV) + p];
      const float pz = pts[2 * NH * (NPQ + NPV) + p];
      const float rx = R00*px + R01*py + R02*pz;  // rel_ori @ pt
      const float ry = R10*px + R11*py + R12*pz;
      const float rz = R20*px + R21*py + R22*pz;
      const float cx = dy*rz - dz*ry;             // direction x rot
      const float cy = dz*rx - dx*rz;
      const float cz = dx*ry - dy*rx;
      if (p < NPQ) {
        const float ex = qpts_sh[0*48 + h*NPQ + p] - cx;
        const float ey = qpts_sh[1*48 + h*NPQ + p] - cy;
        const float ez = qpts_sh[2*48 + h*NPQ + p] - cz;
        ptsum += ex*ex + ey*ey + ez*ez;
      } else {
        vpts_sh[k][h][p - NPQ][0] = cx;
        vpts_sh[k][h][p - NPQ][1] = cy;
        vpts_sh[k][h][p - NPQ][2] = cz;
      }
    }
    const float* qrow = q_sh + h * C_H;
    const float* krow = proj + (size_t)e * PROJW + OFF_KV + h * 2 * C_H;
    float qtk = 0.f;
    #pragma unroll
    for (int c = 0; c < C_H; ++c) qtk += qrow[c] * krow[c];
    float bb = 0.f;
    for (int z = 0; z < C_Z; ++z) bb += z_sh[k][z] * W_b[z * NH + h];
    const float mb = INF_ * (pair_mask[(size_t)n * KK + k] - 1.0f);
    prob[h][k] = qtk * s_qk + bb * s_b - 0.5f * hw_sh[h] * ptsum + mb;
  }
  __syncthreads();

  // ---- softmax over k, per head ----
  if (tid < NH) {
    float mx = -3.4e38f;
    for (int k = 0; k < KK; ++k) mx = fmaxf(mx, prob[tid][k]);
    float ssum = 0.f;
    for (int k = 0; k < KK; ++k) { const float e = expf(prob[tid][k] - mx); prob[tid][k] = e; ssum += e; }
    const float inv = 1.0f / ssum;
    for (int k = 0; k < KK; ++k) prob[tid][k] *= inv;
  }
  __syncthreads();

  float* crow = cat + (size_t)n * CATW;

  // ---- o_pair via WMMA: probs[16x32] @ z[32x128]; wave w -> col groups 2w,2w+1 ----
  {
    const int wave = tid >> 5;
    const int lane = tid & 31;
    const int half = lane >> 4;
    const int l15  = lane & 15;
    #pragma unroll
    for (int g = 0; g < 2; ++g) {
      const int n0 = (wave * 2 + g) * 16;
      v8f acc = {0.f,0.f,0.f,0.f,0.f,0.f,0.f,0.f};
      #pragma unroll
      for (int k0 = 0; k0 < KK; k0 += 4) {
        v2f av, bv;
        av.x = prob[l15][k0 + 2 * half];
        av.y = prob[l15][k0 + 2 * half + 1];
        bv.x = z_sh[k0 + 2 * half][n0 + l15];
        bv.y = z_sh[k0 + 2 * half + 1][n0 + l15];
        acc = wmma4(av, bv, acc);
      }
      #pragma unroll
      for (int r = 0; r < 8; ++r) {
        const int h = r + half * 8;
        if (h < NH) crow[576 + h * C_Z + n0 + l15] = acc[r];
      }
    }
  }

  // ---- o : [H,16] (gathered B operand -> scalar path) ----
  for (int t = tid; t < NH * C_H; t += 128) {
    const int h = t / C_H, c = t % C_H;
    float acc = 0.f;
    for (int k = 0; k < KK; ++k)
      acc += prob[h][k] * proj[(size_t)e_sh[k] * PROJW + OFF_KV + h * 2 * C_H + C_H + c];
    crow[h * C_H + c] = acc;
  }
  // ---- o_pt : [H,PV,3] ----
  for (int t = tid; t < NH * NPV * 3; t += 128) {
    const int h = t / (NPV * 3); const int r = t % (NPV * 3);
    const int p = r / 3, x = r % 3;
    float acc = 0.f;
    for (int k = 0; k < KK; ++k) acc += prob[h][k] * vpts_sh[k][h][p][x];
    opt_sh[h][p][x] = acc;
    crow[NH * C_H + x * (NH * NPV) + h * NPV + p] = acc; // 192 + x*96 + h*8 + p
  }
  __syncthreads();
  // ---- norms : [H,PV] ----
  for (int t = tid; t < NH * NPV; t += 128) {
    const int h = t / NPV, p = t % NPV;
    const float ox = opt_sh[h][p][0], oy = opt_sh[h][p][1], oz = opt_sh[h][p][2];
    crow[480 + h * NPV + p] = sqrtf(ox*ox + oy*oy + oz*oz + EPS_);
  }
}

// ---------------------------------------------------------------
extern "C" void kernel_launch(void* const* d_in, const int* in_sizes, int n_in,
                              void* d_out, int out_size, void* d_ws, size_t ws_size,
                              hipStream_t stream)
{
  const float* s            = (const float*)d_in[0];
  const float* z_e          = (const float*)d_in[1];
  const float* direction    = (const float*)d_in[2];
  const float* rel_ori      = (const float*)d_in[3];
  const float* pair_mask    = (const float*)d_in[4];
  const int*   E_idx        = (const int*)  d_in[5];
  const float* W_q          = (const float*)d_in[6];
  const float* W_kv         = (const float*)d_in[7];
  const float* W_qp         = (const float*)d_in[8];
  const float* W_kvp        = (const float*)d_in[9];
  const float* W_b          = (const float*)d_in[10];
  const float* head_weights = (const float*)d_in[11];
  const float* W_out        = (const float*)d_in[12];
  const float* b_out        = (const float*)d_in[13];

  // workspace layout (fp32): Wcat | proj | cat   (~55 MB total)
  float* ws       = (float*)d_ws;
  float* wcat_buf = ws;                                    // 384*1152
  float* proj_buf = wcat_buf + (size_t)C_N * PROJW;        // 4096*1152
  float* cat_buf  = proj_buf + (size_t)NN * PROJW;         // 4096*2112

  // 1) pack weights
  {
    const int total = C_N * PROJW;
    hipLaunchKernelGGL(pack_w, dim3((total + 255) / 256), dim3(256), 0, stream,
                       W_q, W_kv, W_qp, W_kvp, wcat_buf);
  }
  // 2) fused projection GEMM: s[4096x384] @ Wcat[384x1152]
  hipLaunchKernelGGL(gemm_wmma_f32, dim3((NN / 64) * (PROJW / 64)), dim3(128), 0, stream,
                     s, wcat_buf, (const float*)nullptr, proj_buf, NN, C_N, PROJW);
  // 3) fused attention -> cat
  hipLaunchKernelGGL(ipa_attn, dim3(NN), dim3(128), 0, stream,
                     proj_buf, z_e, direction, rel_ori, pair_mask, E_idx,
                     W_b, head_weights, cat_buf);
  // 4) output GEMM: cat[4096x2112] @ W_out[2112x384] + b_out
  hipLaunchKernelGGL(gemm_wmma_f32, dim3((NN / 64) * (384 / 64)), dim3(128), 0, stream,
                     cat_buf, W_out, b_out, (float*)d_out, NN, CATW, 384);
}